// NCODLoss_1460288881247
// MI455X (gfx1250) — compile-verified
//
#include <hip/hip_runtime.h>
#include <hip/hip_bf16.h>

typedef __attribute__((ext_vector_type(2))) float v2f;
typedef __attribute__((ext_vector_type(8))) float v8f;

#define NCOD_TOTAL_EPOCHS 150
#define NCOD_SPLIT 10

// ---------------------------------------------------------------- utilities

__device__ __forceinline__ int compute_k(int epoch, int PC) {
    if (epoch == 0) return PC;
    float pf = 50.0f + 50.0f * (1.0f - (float)epoch / (float)NCOD_TOTAL_EPOCHS);
    pf = fminf(fmaxf(pf, 1.0f), 100.0f);
    int percent = (int)pf;                 // truncation, like python int()
    int k = (PC * percent) / 100;
    return (k < 1) ? 1 : k;
}

// power-of-two blockDim tree reductions (deterministic order)
__device__ __forceinline__ float rsum(float v, float* s) {
    int t = threadIdx.x, n = blockDim.x;
    s[t] = v; __syncthreads();
    for (int o = n >> 1; o > 0; o >>= 1) { if (t < o) s[t] += s[t + o]; __syncthreads(); }
    float r = s[0]; __syncthreads();
    return r;
}
__device__ __forceinline__ float rmax(float v, float* s) {
    int t = threadIdx.x, n = blockDim.x;
    s[t] = v; __syncthreads();
    for (int o = n >> 1; o > 0; o >>= 1) { if (t < o) s[t] = fmaxf(s[t], s[t + o]); __syncthreads(); }
    float r = s[0]; __syncthreads();
    return r;
}
// exclusive scan over 256 threads (Hillis-Steele, deterministic)
__device__ __forceinline__ int exscan256(int v, int* s) {
    int t = threadIdx.x;
    s[t] = v; __syncthreads();
    #pragma unroll
    for (int o = 1; o < 256; o <<= 1) {
        int x = (t >= o) ? s[t - o] : 0;
        __syncthreads();
        s[t] += x;
        __syncthreads();
    }
    int inc = s[t]; __syncthreads();
    return inc - v;
}

// ---------------------------------------------------------------- kernels

__global__ void k_map_init(int* mapv, int N) {
    int i = blockIdx.x * blockDim.x + threadIdx.x;
    if (i < N) mapv[i] = -1;
}

__global__ void k_map_scatter(int* mapv, const int* __restrict__ indexes, int B) {
    int j = blockIdx.x * blockDim.x + threadIdx.x;
    if (j < B) mapv[indexes[j]] = j;
}

// L2-normalize embeddings -> embN (padded to Mpad rows, pad rows = 0)
__global__ void k_emb_norm(const float* __restrict__ emb, float* __restrict__ embN,
                           int B, int D) {
    __shared__ float red[256];
    int b = blockIdx.x, tid = threadIdx.x;
    int d0 = tid, d1 = tid + 256;
    float x0 = 0.f, x1 = 0.f;
    bool valid = (b < B);
    if (valid) {
        if (d0 < D) x0 = emb[(size_t)b * D + d0];
        if (d1 < D) x1 = emb[(size_t)b * D + d1];
    }
    float ss  = rsum(x0 * x0 + x1 * x1, red);
    float nrm = fmaxf(sqrtf(ss), 1e-12f);
    if (d0 < D) embN[(size_t)b * D + d0] = valid ? x0 / nrm : 0.f;
    if (d1 < D) embN[(size_t)b * D + d1] = valid ? x1 / nrm : 0.f;
}

// Per-class radix-select of k smallest uncertainties + stable compaction.
// One block (256 threads) per class. Dynamic LDS: keys[PC] + hist[256] + scan[256] + misc.
__global__ void k_select(const float* __restrict__ u, const int* __restrict__ bins,
                         int* __restrict__ sel, const int* __restrict__ epoch_p,
                         int C, int PC) {
    extern __shared__ unsigned smem_u[];
    unsigned* keys = smem_u;
    unsigned* hist = keys + PC;
    int*      scan = (int*)(hist + 256);
    int*      misc = scan + 256;

    int c = blockIdx.x, tid = threadIdx.x;
    int k = compute_k(*epoch_p, PC);
    const int cbase = c * PC;

    if (k >= PC) {                         // epoch==0 or full selection
        for (int i = tid; i < PC; i += 256) sel[cbase + i] = bins[cbase + i];
        return;
    }

    // monotone key: flip float bits so unsigned compare == float compare
    for (int i = tid; i < PC; i += 256) {
        int row = bins[cbase + i];
        unsigned bv = __float_as_uint(u[row]);
        keys[i] = bv ^ ((bv >> 31) ? 0xFFFFFFFFu : 0x80000000u);
    }
    __syncthreads();

    unsigned prefix = 0u, maskv = 0u;
    int remaining = k;
    for (int pass = 0; pass < 4; ++pass) {
        int shift = 24 - 8 * pass;
        if (tid < 256) hist[tid] = 0u;
        __syncthreads();
        for (int i = tid; i < PC; i += 256) {
            unsigned key = keys[i];
            if ((key & maskv) == prefix)
                atomicAdd(&hist[(key >> shift) & 255u], 1u);   // LDS ds_add
        }
        __syncthreads();
        if (tid == 0) {
            int cum = 0, d = 0;
            for (; d < 255; ++d) {
                int h = (int)hist[d];
                if (cum + h >= remaining) break;
                cum += h;
            }
            misc[0] = d;
            misc[1] = remaining - cum;
        }
        __syncthreads();
        prefix |= ((unsigned)misc[0]) << shift;
        maskv  |= 0xFFu << shift;
        remaining = misc[1];
        __syncthreads();
    }
    unsigned T   = prefix;          // k-th smallest key
    int neededEq = remaining;       // equals to include (lowest index first)
    int lessTot  = k - neededEq;

    // stable compaction: contiguous per-thread segments + two exclusive scans
    int seg = (PC + 255) / 256;
    int i0 = min(tid * seg, PC), i1 = min(PC, i0 + seg);
    int cl = 0, ceq = 0;
    for (int i = i0; i < i1; ++i) {
        unsigned key = keys[i];
        cl  += (key < T);
        ceq += (key == T);
    }
    int lb = exscan256(cl, scan);
    __syncthreads();
    int eb = exscan256(ceq, scan);
    int lpos = lb, epos = eb;
    for (int i = i0; i < i1; ++i) {
        unsigned key = keys[i];
        if (key < T) {
            sel[cbase + (lpos++)] = bins[cbase + i];
        } else if (key == T) {
            int r = epos++;
            if (r < neededEq) sel[cbase + lessTot + r] = bins[cbase + i];
        }
    }
}

// Partial centroid sums: one block per (class, split); fixed serial row order.
__global__ void k_cent_partial(const float* __restrict__ PE, const float* __restrict__ embN,
                               const int* __restrict__ mapv, const int* __restrict__ sel,
                               const int* __restrict__ epoch_p, float* __restrict__ partial,
                               int C, int PC, int D) {
    int c = blockIdx.x / NCOD_SPLIT;
    int s = blockIdx.x % NCOD_SPLIT;
    int tid = threadIdx.x;
    int k = compute_k(*epoch_p, PC);
    int chunk = (k + NCOD_SPLIT - 1) / NCOD_SPLIT;
    int j0 = s * chunk, j1 = min(k, j0 + chunk);
    int d0 = tid, d1 = tid + 256;
    float a0 = 0.f, a1 = 0.f;
    for (int j = j0; j < j1; ++j) {
        int row = sel[(size_t)c * PC + j];
        int m = mapv[row];
        const float* src = (m >= 0) ? (embN + (size_t)m * D) : (PE + (size_t)row * D);
        if (d0 < D) a0 += src[d0];
        if (d1 < D) a1 += src[d1];
    }
    float* dst = partial + ((size_t)(c * NCOD_SPLIT + s)) * D;
    if (d0 < D) dst[d0] = a0;
    if (d1 < D) dst[d1] = a1;
}

// Combine splits (fixed order), divide by k, L2-normalize -> cent (Npad rows, pad=0)
__global__ void k_cent_norm(const float* __restrict__ partial, float* __restrict__ cent,
                            const int* __restrict__ epoch_p, int C, int PC, int D) {
    __shared__ float red[256];
    int c = blockIdx.x, tid = threadIdx.x;
    int d0 = tid, d1 = tid + 256;
    if (c >= C) {
        if (d0 < D) cent[(size_t)c * D + d0] = 0.f;
        if (d1 < D) cent[(size_t)c * D + d1] = 0.f;
        return;
    }
    int k = compute_k(*epoch_p, PC);
    float m0 = 0.f, m1 = 0.f;
    for (int s = 0; s < NCOD_SPLIT; ++s) {
        const float* p = partial + ((size_t)(c * NCOD_SPLIT + s)) * D;
        if (d0 < D) m0 += p[d0];
        if (d1 < D) m1 += p[d1];
    }
    float invk = 1.0f / (float)k;
    m0 *= invk; m1 *= invk;
    float ss  = rsum(m0 * m0 + m1 * m1, red);
    float nrm = fmaxf(sqrtf(ss), 1e-12f);
    if (d0 < D) cent[(size_t)c * D + d0] = m0 / nrm;
    if (d1 < D) cent[(size_t)c * D + d1] = m1 / nrm;
}

// gemm[b][n] = sum_d embN[b][d] * cent[n][d] via V_WMMA_F32_16X16X4_F32.
// One 16x16 output tile per wave32; A/B fp32 16x4 layout (lane half = K pair).
__global__ void k_gemm_wmma(const float* __restrict__ A, const float* __restrict__ Bm,
                            float* __restrict__ Cm, int D, int Npad, int Ntiles, int nTilesTotal) {
    int wid  = (blockIdx.x * blockDim.x + threadIdx.x) >> 5;   // wave-uniform
    if (wid >= nTilesTotal) return;
    int lane = threadIdx.x & 31;
    int mt = wid / Ntiles, nt = wid % Ntiles;
    int half = lane >> 4, l16 = lane & 15;

    const float* arow = A  + (size_t)(mt * 16 + l16) * D;
    const float* brow = Bm + (size_t)(nt * 16 + l16) * D;

    v8f acc = {};
    int k4 = D & ~3;
    for (int k0 = 0; k0 < k4; k0 += 4) {
        v2f a = *(const v2f*)(arow + k0 + half * 2);
        v2f b = *(const v2f*)(brow + k0 + half * 2);
        acc = __builtin_amdgcn_wmma_f32_16x16x4_f32(
            false, a, false, b, (short)0, acc, false, false);
    }
    if (k4 < D) {                           // K tail (zero-padded)
        int kk = k4 + half * 2;
        v2f a = { (kk < D) ? arow[kk] : 0.f, (kk + 1 < D) ? arow[kk + 1] : 0.f };
        v2f b = { (kk < D) ? brow[kk] : 0.f, (kk + 1 < D) ? brow[kk + 1] : 0.f };
        acc = __builtin_amdgcn_wmma_f32_16x16x4_f32(
            false, a, false, b, (short)0, acc, false, false);
    }
    // D-matrix layout: VGPR r -> row M = r + half*8, col = lane%16
    int row0 = mt * 16 + half * 8;
    int col  = nt * 16 + l16;
    float* out = Cm + (size_t)row0 * Npad + col;
    #pragma unroll
    for (int r = 0; r < 8; ++r) out[(size_t)r * Npad] = acc[r];
}

// Per-row loss pieces: hard-CE term, soft-CE term, consistency term.
__global__ void k_rowloss(const float* __restrict__ gemm, const float* __restrict__ logits,
                          const float* __restrict__ u, const int* __restrict__ indexes,
                          const int* __restrict__ targets, float* __restrict__ rowOut,
                          int B, int C, int Npad) {
    extern __shared__ float sm[];
    float* sl  = sm;            // [C] soft labels
    float* aj  = sm + C;        // [C] scratch (probs numer, then adjusted)
    float* red = sm + 2 * C;    // [blockDim]
    int b = blockIdx.x, tid = threadIdx.x, nthr = blockDim.x;

    // soft_labels = softmax(gemm row)
    float mx = -3.4e38f;
    for (int c = tid; c < C; c += nthr) mx = fmaxf(mx, gemm[(size_t)b * Npad + c]);
    mx = rmax(mx, red);
    float se = 0.f;
    for (int c = tid; c < C; c += nthr) {
        float e = expf(gemm[(size_t)b * Npad + c] - mx);
        sl[c] = e; se += e;
    }
    se = rsum(se, red);
    for (int c = tid; c < C; c += nthr) sl[c] /= se;
    __syncthreads();

    // probs = softmax(logits row)
    float mx2 = -3.4e38f;
    for (int c = tid; c < C; c += nthr) mx2 = fmaxf(mx2, logits[(size_t)b * C + c]);
    mx2 = rmax(mx2, red);
    float se2 = 0.f;
    for (int c = tid; c < C; c += nthr) {
        float e = expf(logits[(size_t)b * C + c] - mx2);
        aj[c] = e; se2 += e;
    }
    se2 = rsum(se2, red);

    float uv = 1.0f / (1.0f + expf(-u[indexes[b]]));

    // adjusted = max(probs + uv*soft_labels, eps); renormalize
    float sadj = 0.f;
    for (int c = tid; c < C; c += nthr) {
        float p = aj[c] / se2;
        float a = fmaxf(p + uv * sl[c], 1e-6f);
        aj[c] = a; sadj += a;
    }
    sadj = rsum(sadj, red);

    float softacc = 0.f, consacc = 0.f;
    for (int c = tid; c < C; c += nthr) {
        float a = aj[c] / sadj;
        float s = sl[c];
        softacc += s * logf(a);
        float dd = a - s;
        consacc += dd * dd;
    }
    softacc = rsum(softacc, red);
    consacc = rsum(consacc, red);

    if (tid == 0) {
        int tgt = targets[b];
        float lp = (logits[(size_t)b * C + tgt] - mx2) - logf(se2);  // log_softmax
        float ce = -lp;
        rowOut[(size_t)b * 3 + 0] = (1.0f - uv) * ce;
        rowOut[(size_t)b * 3 + 1] = -softacc;
        rowOut[(size_t)b * 3 + 2] = consacc;
    }
}

__global__ void k_final(const float* __restrict__ rowOut, float* __restrict__ out, int B, int C) {
    __shared__ float red[256];
    int tid = threadIdx.x;
    float h = 0.f, s = 0.f, co = 0.f;
    for (int b = tid; b < B; b += 256) {
        h  += rowOut[(size_t)b * 3 + 0];
        s  += rowOut[(size_t)b * 3 + 1];
        co += rowOut[(size_t)b * 3 + 2];
    }
    h  = rsum(h, red);
    s  = rsum(s, red);
    co = rsum(co, red);
    if (tid == 0)
        out[0] = h / (float)B + s / (float)B + co / ((float)B * (float)C);
}

// ---------------------------------------------------------------- launcher

extern "C" void kernel_launch(void* const* d_in, const int* in_sizes, int n_in,
                              void* d_out, int out_size, void* d_ws, size_t ws_size,
                              hipStream_t stream) {
    const float* logits     = (const float*)d_in[0];
    const float* embeddings = (const float*)d_in[1];
    const float* u          = (const float*)d_in[2];
    const float* past       = (const float*)d_in[3];
    const int*   indexes    = (const int*)d_in[4];
    const int*   targets    = (const int*)d_in[5];
    const int*   bins       = (const int*)d_in[6];
    const int*   epoch_p    = (const int*)d_in[7];

    const int B  = in_sizes[5];               // targets
    const int C  = in_sizes[0] / B;           // logits [B,C]
    const int D  = in_sizes[1] / B;           // embeddings [B,D]
    const int N  = in_sizes[2];               // u [N]
    const int PC = in_sizes[6] / C;           // bins [C,PC]
    const int Mpad = ((B + 15) / 16) * 16;
    const int Npad = ((C + 15) / 16) * 16;

    // workspace carve (256B aligned)
    char* w = (char*)d_ws;
    auto carve = [&](size_t bytes) -> void* {
        void* p = (void*)w;
        w += (bytes + 255) & ~(size_t)255;
        return p;
    };
    float* embN    = (float*)carve((size_t)Mpad * D * 4);
    float* cent    = (float*)carve((size_t)Npad * D * 4);
    float* gemm    = (float*)carve((size_t)Mpad * Npad * 4);
    float* partial = (float*)carve((size_t)C * NCOD_SPLIT * D * 4);
    float* rowOut  = (float*)carve((size_t)B * 3 * 4);
    int*   mapv    = (int*)carve((size_t)N * 4);
    int*   sel     = (int*)carve((size_t)C * PC * 4);
    (void)ws_size;

    // 1) row -> batch-position map for the scatter pe[indexes] = emb
    k_map_init<<<(N + 255) / 256, 256, 0, stream>>>(mapv, N);
    k_map_scatter<<<(B + 255) / 256, 256, 0, stream>>>(mapv, indexes, B);

    // 2) normalize embeddings (padded to Mpad rows)
    k_emb_norm<<<Mpad, 256, 0, stream>>>(embeddings, embN, B, D);

    // 3) per-class k-smallest-uncertainty selection (device-side k from epoch)
    size_t selLds = ((size_t)PC + 256 + 256 + 8) * 4;
    k_select<<<C, 256, selLds, stream>>>(u, bins, sel, epoch_p, C, PC);

    // 4) centroid partial sums (deterministic split), combine + normalize
    k_cent_partial<<<C * NCOD_SPLIT, 256, 0, stream>>>(past, embN, mapv, sel, epoch_p,
                                                       partial, C, PC, D);
    k_cent_norm<<<Npad, 256, 0, stream>>>(partial, cent, epoch_p, C, PC, D);

    // 5) emb @ cent^T via fp32 WMMA (one 16x16 tile per wave32)
    int tiles = (Mpad / 16) * (Npad / 16);
    int gemmBlocks = (tiles * 32 + 255) / 256;
    k_gemm_wmma<<<gemmBlocks, 256, 0, stream>>>(embN, cent, gemm, D, Npad, Npad / 16, tiles);

    // 6) rowwise loss pieces, final reduction
    size_t rlLds = ((size_t)2 * C + 128 + 8) * 4;
    k_rowloss<<<B, 128, rlLds, stream>>>(gemm, logits, u, indexes, targets, rowOut, B, C, Npad);
    k_final<<<1, 256, 0, stream>>>(rowOut, (float*)d_out, B, C);
}